// SparseArch_10299331576392
// MI455X (gfx1250) — compile-verified
//
#include <hip/hip_runtime.h>
#include <stdint.h>

// ---------------------------------------------------------------------------
// TBE pooled-embedding forward for MI455X (gfx1250).
// Memory-bound random gather: CDNA5 async global->LDS DMA path
// (GLOBAL_LOAD_ASYNC_TO_LDS_B128 / ASYNCcnt). Each wave keeps 20 row fetches
// in flight with zero VGPR landing buffers; in-order async completion lets us
// split the wait (wait<=10, read half, wait 0, read rest) to overlap LDS
// accumulation with outstanding HBM fetches.
// ---------------------------------------------------------------------------

typedef float v4f __attribute__((ext_vector_type(4)));

constexpr int kT             = 16;      // tables
constexpr int kE             = 100000;  // rows per table
constexpr int kD             = 128;     // embedding dim (512 B per row)
constexpr int kRowBytes      = kD * 4;
constexpr int kMaxChunk      = 20;      // rows staged per wave per wait (L == 20)
constexpr int kWavesPerBlock = 4;       // 4 waves * 20 rows * 512 B = 40 KB LDS/block
constexpr int kThreads       = kWavesPerBlock * 32;

__device__ __forceinline__ void async_row_to_lds(uint32_t laddr, uint64_t gaddr) {
    asm volatile("global_load_async_to_lds_b128 %0, %1, off"
                 :
                 : "v"(laddr), "v"(gaddr)
                 : "memory");
}

__global__ __launch_bounds__(kThreads)
void tbe_pooled_fwd(const float* __restrict__ weights,   // [T, E, D]
                    const int*   __restrict__ indices,   // [T*B*L]
                    const int*   __restrict__ offsets,   // [T*B + 1]
                    float*       __restrict__ out,       // [B, T*D]
                    int num_bags, int bags_per_table, int table_shift)
{
    __shared__ __align__(16) float stage[kWavesPerBlock][kMaxChunk][kD];

    const int lane = threadIdx.x & 31;
    const int wave = threadIdx.x >> 5;
    const int bag  = blockIdx.x * kWavesPerBlock + wave;   // table-major bag id
    if (bag >= num_bags) return;

    // B is a power of two in this workload: shift instead of integer divide.
    const int t     = (table_shift >= 0) ? (bag >> table_shift) : (bag / bags_per_table);
    const int batch = bag - t * bags_per_table;

    const int start = offsets[bag];
    const int end   = offsets[bag + 1];

    // Wave-uniform base address of this bag's table.
    const uint64_t table_base =
        (uint64_t)(uintptr_t)weights + (uint64_t)t * (uint64_t)kE * (uint64_t)kRowBytes;
    const uint32_t lane_byte = (uint32_t)lane * 16u;   // each lane covers 16 B of a row
    const uint32_t lds_base  =
        (uint32_t)(uintptr_t)&stage[wave][0][lane * 4]; // low 32 bits = LDS offset

    v4f acc = (v4f)0.0f;

    for (int cbase = start; cbase < end; cbase += kMaxChunk) {
        const int chunk = min(end - cbase, kMaxChunk);

        // One coalesced index load per lane; v_readlane broadcasts keep row
        // address math wave-uniform.
        int myidx = 0;
        if (lane < chunk) myidx = indices[cbase + lane];

        if (chunk == kMaxChunk) {
            // --- Fast path: fully unrolled, 20 back-to-back async issues ---
#pragma unroll
            for (int i = 0; i < kMaxChunk; ++i) {
                const uint32_t row = (uint32_t)__builtin_amdgcn_readlane(myidx, i);
                async_row_to_lds(lds_base + (uint32_t)i * kRowBytes,
                                 table_base + (uint64_t)row * kRowBytes + lane_byte);
            }
            // Async loads complete in order: rows 0..9 are done once <=10 remain.
            asm volatile("s_wait_asynccnt 0xa" ::: "memory");
#pragma unroll
            for (int i = 0; i < kMaxChunk / 2; ++i)
                acc += *(const v4f*)&stage[wave][i][lane * 4];
            asm volatile("s_wait_asynccnt 0x0" ::: "memory");
#pragma unroll
            for (int i = kMaxChunk / 2; i < kMaxChunk; ++i)
                acc += *(const v4f*)&stage[wave][i][lane * 4];
        } else {
            // --- General (ragged tail) path ---
            for (int i = 0; i < chunk; ++i) {
                const uint32_t row = (uint32_t)__builtin_amdgcn_readlane(myidx, i);
                async_row_to_lds(lds_base + (uint32_t)i * kRowBytes,
                                 table_base + (uint64_t)row * kRowBytes + lane_byte);
            }
            asm volatile("s_wait_asynccnt 0x0" ::: "memory");
            for (int i = 0; i < chunk; ++i)
                acc += *(const v4f*)&stage[wave][i][lane * 4];
        }
    }

    // Output is write-once streaming data: non-temporal so the 192 MB L2 stays
    // dedicated to embedding-table reuse.
    float* op = out + (size_t)batch * (size_t)(kT * kD) + (size_t)t * kD + (size_t)lane * 4;
    __builtin_nontemporal_store(acc, (v4f*)op);
}

extern "C" void kernel_launch(void* const* d_in, const int* in_sizes, int n_in,
                              void* d_out, int out_size, void* d_ws, size_t ws_size,
                              hipStream_t stream)
{
    const float* weights = (const float*)d_in[0];
    const int*   indices = (const int*)d_in[1];
    const int*   offsets = (const int*)d_in[2];
    float*       out     = (float*)d_out;

    const int num_bags       = in_sizes[2] - 1;   // T*B
    const int bags_per_table = num_bags / kT;     // B

    // log2(B) when B is a power of two, else -1 (kernel falls back to divide).
    int table_shift = -1;
    if (bags_per_table > 0 && (bags_per_table & (bags_per_table - 1)) == 0) {
        table_shift = 0;
        while ((1 << table_shift) < bags_per_table) ++table_shift;
    }

    const int blocks = (num_bags + kWavesPerBlock - 1) / kWavesPerBlock;
    hipLaunchKernelGGL(tbe_pooled_fwd, dim3(blocks), dim3(kThreads), 0, stream,
                       weights, indices, offsets, out, num_bags, bags_per_table,
                       table_shift);
}